// SSIM_14199161881041
// MI455X (gfx1250) — compile-verified
//
#include <hip/hip_runtime.h>

// SSIM (11-tap separable Gaussian, zero 'same' padding) for 16x1x1024x1024 fp32.
// Fused single pass: block-shared halo in LDS, blurs via V_WMMA_F32_16X16X4_F32
// (OUT = Wv x IN x Wh with banded weight matrices), deterministic two-level reduce.

typedef float v2f __attribute__((ext_vector_type(2)));
typedef float v8f __attribute__((ext_vector_type(8)));

#define WSZ 11
#define C1F 1.0e-4f      // 0.01^2
#define C2F 9.0e-4f      // 0.03^2

#define IMG_H 1024
#define IMG_W 1024
#define N_IMG 16

// Block tile: 32 rows x 64 cols = 8 subtiles of 16x16; 4 waves x 2 subtiles each.
#define WAVES_PER_BLOCK 4
#define BLOCK_THREADS (WAVES_PER_BLOCK * 32)
#define BT_H 32
#define BT_W 64
#define SUBTILES 8
#define SUB_PER_WAVE 2
#define BLOCKS_PER_IMG ((IMG_H / BT_H) * (IMG_W / BT_W))   // 32 * 16 = 512
#define NUM_BLOCKS (N_IMG * BLOCKS_PER_IMG)                // 8192

// Shared halo: (BT_H+10) x (BT_W+10) = 42x74, padded to 44x80.
// Pad rows/cols only ever see zero band weights (window-relative K rows 26..27
// and T columns 26..27 have w[j]=0 for j>10), so garbage there is harmless.
#define HALO_H 44
#define HALO_W 80
#define HALO_ELEMS (HALO_H * HALO_W)                       // 3520

// Per-wave T scratch: 5 planes x 16 rows x stride 28 (cols >=28 never read).
#define T_STRIDE 28
#define T_PLANE (16 * T_STRIDE)                            // 448
#define T_WAVE (5 * T_PLANE)                               // 2240

#define HALO1_OFF 0
#define HALO2_OFF HALO_ELEMS
#define T_OFF (2 * HALO_ELEMS)                             // 7040
#define POOL_FLOATS (T_OFF + WAVES_PER_BLOCK * T_WAVE)     // 16000 -> 64000 B

__global__ __launch_bounds__(BLOCK_THREADS)
void ssim_tile_kernel(const float* __restrict__ img1,
                      const float* __restrict__ img2,
                      const float* __restrict__ window,
                      float* __restrict__ partial)
{
    __shared__ float pool[POOL_FLOATS];
    __shared__ float wtab[12];                 // wtab[11] == 0 sentinel
    __shared__ float red[WAVES_PER_BLOCK];

    const int tid  = threadIdx.x;
    const int wave = tid >> 5;
    const int lane = tid & 31;

    if (tid < 12) {
        float wv = 0.0f;
        if (tid < WSZ) wv = window[tid];
        wtab[tid] = wv;
    }

    // ---- which 32x64 block tile this block owns ----
    const int b    = blockIdx.x;
    const int img  = b / BLOCKS_PER_IMG;
    const int rem  = b - img * BLOCKS_PER_IMG;
    const int brow = (rem / (IMG_W / BT_W)) * BT_H;        // (rem/16)*32
    const int bcol = (rem % (IMG_W / BT_W)) * BT_W;        // (rem%16)*64
    const float* base1 = img1 + (size_t)img * (IMG_H * IMG_W);
    const float* base2 = img2 + (size_t)img * (IMG_H * IMG_W);

    // ---- cooperative stage of the shared 44x80 halo of both images ----
    float* H1 = &pool[HALO1_OFF];
    float* H2 = &pool[HALO2_OFF];
    for (int idx = tid; idx < HALO_ELEMS; idx += BLOCK_THREADS) {
        const int r  = idx / HALO_W;
        const int c  = idx - r * HALO_W;
        const int gr = brow + r - 5;
        const int gc = bcol + c - 5;
        const bool ok = (gr >= 0) & (gr < IMG_H) & (gc >= 0) & (gc < IMG_W);
        float a = 0.0f, bb = 0.0f;
        if (ok) {
            a  = base1[gr * IMG_W + gc];
            bb = base2[gr * IMG_W + gc];
        }
        H1[idx] = a;
        H2[idx] = bb;
    }
    __syncthreads();

    const int half = lane >> 4;    // K-split half of WMMA operands
    const int m    = lane & 15;    // matrix row (A) / col (B)

    // Band weight: w[j] for j in [0,10], else 0 (negative j wraps unsigned).
    auto bw = [&](int j) -> float {
        unsigned u = (unsigned)j;
        return wtab[u < 11u ? u : 11u];
    };

    const v8f vzero = {0.f,0.f,0.f,0.f,0.f,0.f,0.f,0.f};
    float* Tw = &pool[T_OFF + wave * T_WAVE];   // private per-wave T scratch

    float lsum = 0.0f;

    for (int i = 0; i < SUB_PER_WAVE; ++i) {
        const int s    = wave * SUB_PER_WAVE + i;  // subtile 0..7 (2x4 grid)
        const int srow = (s >> 2) * 16;            // 0 or 16
        const int scol = (s & 3) * 16;             // 0,16,32,48
        const float* IN1 = H1 + srow * HALO_W + scol;
        const float* IN2 = H2 + srow * HALO_W + scol;

        // ===== phase 1: vertical blur, T_p = Wv(16x28) x IN_p(28x32) =====
        v8f accT[5][2];
#pragma unroll
        for (int p = 0; p < 5; ++p) { accT[p][0] = vzero; accT[p][1] = vzero; }

#pragma unroll
        for (int k = 0; k < 7; ++k) {
            const int kr = 4 * k + 2 * half;
            v2f a; a.x = bw(kr - m); a.y = bw(kr + 1 - m);   // Wv[m][kr], Wv[m][kr+1]
#pragma unroll
            for (int g = 0; g < 2; ++g) {
                const int col = 16 * g + m;
                const float x0 = IN1[(kr    ) * HALO_W + col];
                const float x1 = IN1[(kr + 1) * HALO_W + col];
                const float y0 = IN2[(kr    ) * HALO_W + col];
                const float y1 = IN2[(kr + 1) * HALO_W + col];
                v2f b0; b0.x = x0;      b0.y = x1;
                v2f b1; b1.x = y0;      b1.y = y1;
                v2f b2; b2.x = x0 * x0; b2.y = x1 * x1;
                v2f b3; b3.x = y0 * y0; b3.y = y1 * y1;
                v2f b4; b4.x = x0 * y0; b4.y = x1 * y1;
                accT[0][g] = __builtin_amdgcn_wmma_f32_16x16x4_f32(false, a, false, b0, (short)0, accT[0][g], false, false);
                accT[1][g] = __builtin_amdgcn_wmma_f32_16x16x4_f32(false, a, false, b1, (short)0, accT[1][g], false, false);
                accT[2][g] = __builtin_amdgcn_wmma_f32_16x16x4_f32(false, a, false, b2, (short)0, accT[2][g], false, false);
                accT[3][g] = __builtin_amdgcn_wmma_f32_16x16x4_f32(false, a, false, b3, (short)0, accT[3][g], false, false);
                accT[4][g] = __builtin_amdgcn_wmma_f32_16x16x4_f32(false, a, false, b4, (short)0, accT[4][g], false, false);
            }
        }

        // ---- spill T (C/D layout) to LDS; reload in A layout for phase 2 ----
        // T columns >= 28 are never read (horizontal K stops at 27): skip them
        // so the stride-28 packing cannot wrap into the next row.
#pragma unroll
        for (int p = 0; p < 5; ++p) {
            float* T = Tw + p * T_PLANE;
#pragma unroll
            for (int g = 0; g < 2; ++g) {
                const int col = 16 * g + m;
                if (col < T_STRIDE) {
#pragma unroll
                    for (int v = 0; v < 8; ++v)
                        T[(v + 8 * half) * T_STRIDE + col] = accT[p][g][v];
                }
            }
        }
        __syncthreads();   // uniform; also orders same-wave T write->read

        // ===== phase 2: horizontal blur, OUT_p = T_p(16x28) x Wh(28x16) =====
        v8f accO[5];
#pragma unroll
        for (int p = 0; p < 5; ++p) accO[p] = vzero;

#pragma unroll
        for (int k = 0; k < 7; ++k) {
            const int kr = 4 * k + 2 * half;
            v2f bv; bv.x = bw(kr - m); bv.y = bw(kr + 1 - m);  // Wh[kr][m], Wh[kr+1][m]
#pragma unroll
            for (int p = 0; p < 5; ++p) {
                const float* T = Tw + p * T_PLANE;
                v2f a; a.x = T[m * T_STRIDE + kr]; a.y = T[m * T_STRIDE + kr + 1];
                accO[p] = __builtin_amdgcn_wmma_f32_16x16x4_f32(false, a, false, bv, (short)0, accO[p], false, false);
            }
        }
        __syncthreads();   // uniform; T region reused next iteration

        // ===== SSIM map for the 8 pixels this lane holds =====
#pragma unroll
        for (int v = 0; v < 8; ++v) {
            const float mu1 = accO[0][v], mu2 = accO[1][v];
            const float ex2 = accO[2][v], ey2 = accO[3][v], exy = accO[4][v];
            const float mu1s = mu1 * mu1, mu2s = mu2 * mu2, m12 = mu1 * mu2;
            const float s1 = ex2 - mu1s, s2 = ey2 - mu2s, s12 = exy - m12;
            const float num = (2.0f * m12 + C1F) * (2.0f * s12 + C2F);
            const float den = (mu1s + mu2s + C1F) * (s1 + s2 + C2F);
            lsum += num / den;
        }
    }

    // ---- deterministic reduction: wave shuffle -> block partial ----
#pragma unroll
    for (int off = 16; off >= 1; off >>= 1)
        lsum += __shfl_xor(lsum, off, 32);
    if (lane == 0) red[wave] = lsum;
    __syncthreads();
    if (tid == 0) {
        float ssum = 0.0f;
#pragma unroll
        for (int wv = 0; wv < WAVES_PER_BLOCK; ++wv) ssum += red[wv];
        partial[blockIdx.x] = ssum;
    }
}

__global__ __launch_bounds__(256)
void ssim_reduce_kernel(const float* __restrict__ partial, int n,
                        float* __restrict__ out)
{
    __shared__ float sh[256];
    float s = 0.0f;
    for (int i = threadIdx.x; i < n; i += 256) s += partial[i];
    sh[threadIdx.x] = s;
    __syncthreads();
    for (int off = 128; off >= 1; off >>= 1) {
        if ((int)threadIdx.x < off) sh[threadIdx.x] += sh[threadIdx.x + off];
        __syncthreads();
    }
    if (threadIdx.x == 0)
        out[0] = sh[0] * (1.0f / (16.0f * 1024.0f * 1024.0f));
}

extern "C" void kernel_launch(void* const* d_in, const int* in_sizes, int n_in,
                              void* d_out, int out_size, void* d_ws, size_t ws_size,
                              hipStream_t stream) {
    const float* img1   = (const float*)d_in[0];
    const float* img2   = (const float*)d_in[1];
    const float* window = (const float*)d_in[2];
    float* partial = (float*)d_ws;          // NUM_BLOCKS floats = 32 KB scratch
    float* out     = (float*)d_out;

    ssim_tile_kernel<<<NUM_BLOCKS, BLOCK_THREADS, 0, stream>>>(img1, img2, window, partial);
    ssim_reduce_kernel<<<1, 256, 0, stream>>>(partial, NUM_BLOCKS, out);
}